// LinearKernel_58720792871620
// MI455X (gfx1250) — compile-verified
//
#include <hip/hip_runtime.h>
#include <cstddef>
#include <cstdint>

typedef __attribute__((ext_vector_type(2))) float v2f;
typedef __attribute__((ext_vector_type(8))) float v8f;

#define TILE  128           // C tile per workgroup (128x128)
#define KT    32            // K chunk staged in LDS
#define LDSTR 36            // dwords/row: 144B = 16B-aligned AND conflict-free
#define BUFSZ (TILE * LDSTR)  // dwords per matrix per stage

// Async global->LDS DMA (CDNA5): VDST = LDS byte address VGPR, VADDR = 64-bit
// global address VGPR pair. Tracked by ASYNCcnt.
__device__ __forceinline__ void async_copy_b128(uint32_t lds_byte_off,
                                                const float* gptr) {
  asm volatile("global_load_async_to_lds_b128 %0, %1, off"
               :: "v"(lds_byte_off), "v"(gptr)
               : "memory");
}

#define S_WAIT_ASYNCCNT(N) asm volatile("s_wait_asynccnt " #N ::: "memory")

__global__ __launch_bounds__(256)
void wmma_gemm_nt_f32(const float* __restrict__ A,
                      const float* __restrict__ B,
                      float* __restrict__ C,
                      int N, int M, int D) {
  // [stage0: A | B][stage1: A | B]
  __shared__ float lds[4 * BUFSZ];

  const int tid  = threadIdx.x;
  const int lane = tid & 31;
  const int wave = tid >> 5;
  const int wm   = wave & 3;   // 4 row-bands of 32
  const int wn   = wave >> 2;  // 2 col-bands of 64

  const long long rowBase = (long long)blockIdx.y * TILE;
  const long long colBase = (long long)blockIdx.x * TILE;

  // WMMA f32 16x16x4 fragment decode: row/col = lane&15, K pair = 2*(lane>>4)
  const int fr = lane & 15;
  const int fk = (lane >> 4) << 1;

  // global->LDS staging map: thread covers rows tid>>3 + p*32, cols (tid&7)*4
  const int ldRow = tid >> 3;
  const int ldCol = (tid & 7) << 2;

  const float* Ag = A + (rowBase + ldRow) * (long long)D + ldCol;
  const float* Bg = B + (colBase + ldRow) * (long long)D + ldCol;

  // LDS byte addresses (per stage) for this thread's first staging slot
  uint32_t ldsA[2], ldsB[2];
  #pragma unroll
  for (int s = 0; s < 2; ++s) {
    ldsA[s] = (uint32_t)(size_t)&lds[s * 2 * BUFSZ + ldRow * LDSTR + ldCol];
    ldsB[s] = (uint32_t)(size_t)&lds[s * 2 * BUFSZ + BUFSZ + ldRow * LDSTR + ldCol];
  }

  const v8f vzero = {};
  v8f acc[2][4];
  #pragma unroll
  for (int i = 0; i < 2; ++i)
    #pragma unroll
    for (int j = 0; j < 4; ++j)
      acc[i][j] = vzero;

  // Prologue: stage chunk 0 into buffer 0 (8 async b128 per thread)
  #pragma unroll
  for (int p = 0; p < 4; ++p) {
    async_copy_b128(ldsA[0] + p * 32 * LDSTR * 4, Ag + (long long)p * 32 * D);
    async_copy_b128(ldsB[0] + p * 32 * LDSTR * 4, Bg + (long long)p * 32 * D);
  }

  for (int k0 = 0; k0 < D; k0 += KT) {
    const int s = (k0 / KT) & 1;

    if (k0 + KT < D) {
      // Prefetch next chunk into the other stage while this one computes
      const int ns = s ^ 1;
      #pragma unroll
      for (int p = 0; p < 4; ++p) {
        async_copy_b128(ldsA[ns] + p * 32 * LDSTR * 4,
                        Ag + (long long)p * 32 * D + (k0 + KT));
        async_copy_b128(ldsB[ns] + p * 32 * LDSTR * 4,
                        Bg + (long long)p * 32 * D + (k0 + KT));
      }
      // 16 outstanding; loads retire in order -> <=8 means chunk k is in LDS
      S_WAIT_ASYNCCNT(8);
    } else {
      S_WAIT_ASYNCCNT(0);
    }
    __syncthreads();

    const float* la = &lds[s * 2 * BUFSZ];
    const float* lb = la + BUFSZ;

    #pragma unroll
    for (int kk = 0; kk < KT; kk += 4) {
      v2f af[2], bf[4];
      #pragma unroll
      for (int i = 0; i < 2; ++i)
        af[i] = *(const v2f*)&la[(wm * 32 + i * 16 + fr) * LDSTR + kk + fk];
      #pragma unroll
      for (int j = 0; j < 4; ++j)
        bf[j] = *(const v2f*)&lb[(wn * 64 + j * 16 + fr) * LDSTR + kk + fk];

      #pragma unroll
      for (int i = 0; i < 2; ++i)
        #pragma unroll
        for (int j = 0; j < 4; ++j)
          acc[i][j] = __builtin_amdgcn_wmma_f32_16x16x4_f32(
              false, af[i], false, bf[j], (short)0, acc[i][j], false, false);
    }
    __syncthreads();  // release this stage for refill at iteration k+1
  }

  // C/D layout: VGPR g -> row g + 8*(lane>>4), col lane&15
  const int crow0 = (lane >> 4) << 3;
  const int ccol  = lane & 15;
  #pragma unroll
  for (int i = 0; i < 2; ++i)
    #pragma unroll
    for (int j = 0; j < 4; ++j) {
      const long long row = rowBase + wm * 32 + i * 16 + crow0;
      const long long col = colBase + wn * 64 + j * 16 + ccol;
      float* cp = C + row * (long long)M + col;
      #pragma unroll
      for (int g = 0; g < 8; ++g)
        cp[(long long)g * M] = acc[i][j][g];
    }
}

extern "C" void kernel_launch(void* const* d_in, const int* in_sizes, int n_in,
                              void* d_out, int out_size, void* d_ws, size_t ws_size,
                              hipStream_t stream) {
  const float* A = (const float*)d_in[0];  // [N, D] fp32
  const float* B = (const float*)d_in[1];  // [M, D] fp32
  float* C = (float*)d_out;                // [N, M] fp32

  const int D = 4096;
  const int N = in_sizes[0] / D;
  const int M = in_sizes[1] / D;

  dim3 grid(M / TILE, N / TILE);  // 32 x 32 workgroups
  wmma_gemm_nt_f32<<<grid, dim3(256), 0, stream>>>(A, B, C, N, M, D);
}